// MultiLabelsGaussianFocalLoss_39874476376216
// MI455X (gfx1250) — compile-verified
//
#include <hip/hip_runtime.h>

// MultiLabelsGaussianFocalLoss for MI455X (gfx1250, wave32).
// pred, target: 8192x8192 f32. Output: scalar f32 mean loss.
//
// loss(i,j) = pos ? -log(1-|t-p|)*|t-p|^2
//                 : -log(1-p+eps)*p^2*(1-t)^4
// pos = 3x3 max-dilation of (target==1.0)  (0-padded 'SAME' == index clamp)

typedef __attribute__((ext_vector_type(2))) float v2f;
typedef __attribute__((ext_vector_type(8))) float v8f;

#define HH 8192
#define WW 8192
#define BAND 16            // rows per workgroup
#define TPB 256            // 8 wave32s
#define QPR (WW / 4)       // float4 quads per row = 2048
#define KITER (QPR / TPB)  // 8 quads per thread per row
#define NBLOCKS (HH / BAND) // 512

__device__ __forceinline__ float loss_elem(float p, float tv, float m9) {
    // single-log formulation: select log argument and multiplier by mask
    const bool pos = (m9 == 1.0f);
    const float d    = fabsf(tv - p);
    const float xpos = 1.0f - d;                 // > 0 since d < 1
    const float xneg = 1.0f - p + 1e-12f;        // p in (0.01, 0.99)
    const float omt  = 1.0f - tv;
    const float omt2 = omt * omt;
    const float apos = d * d;
    const float aneg = p * p * omt2 * omt2;
    const float x = pos ? xpos : xneg;
    const float a = pos ? apos : aneg;
    // -log(x)*a ; __logf -> v_log_f32 (TRANS pipe, co-executes with VALU)
    return -__logf(x) * a;
}

__global__ void __launch_bounds__(TPB)
focal_main(const float* __restrict__ pred,
           const float* __restrict__ targ,
           float* __restrict__ partial) {
    const int row0 = blockIdx.x * BAND;
    const int t = threadIdx.x;

    float acc = 0.0f;

    for (int r = 0; r < BAND; ++r) {
        const int i  = row0 + r;
        const int im = (i == 0)      ? 0        : i - 1;
        const int ip = (i == HH - 1) ? (HH - 1) : i + 1;
        const float* __restrict__ tm = targ + (size_t)im * WW;
        const float* __restrict__ tc = targ + (size_t)i  * WW;
        const float* __restrict__ tp = targ + (size_t)ip * WW;
        const float* __restrict__ pr = pred + (size_t)i  * WW;

        for (int k = 0; k < KITER; ++k) {
            const int j = (t + k * TPB) * 4;   // column of this float4

            const float4 p4 = *(const float4*)(pr + j);
            const float4 a4 = *(const float4*)(tm + j);
            const float4 b4 = *(const float4*)(tc + j);
            const float4 c4 = *(const float4*)(tp + j);

            // vertical 3-row max per column (v_max3_f32)
            const float vm0 = fmaxf(fmaxf(a4.x, b4.x), c4.x);
            const float vm1 = fmaxf(fmaxf(a4.y, b4.y), c4.y);
            const float vm2 = fmaxf(fmaxf(a4.z, b4.z), c4.z);
            const float vm3 = fmaxf(fmaxf(a4.w, b4.w), c4.w);

            // halo columns (clamped), served from cache
            const int jl = (j == 0) ? 0 : j - 1;
            const int jr = (j + 4 >= WW) ? (WW - 1) : j + 4;
            const float vml = fmaxf(fmaxf(tm[jl], tc[jl]), tp[jl]);
            const float vmr = fmaxf(fmaxf(tm[jr], tc[jr]), tp[jr]);

            // horizontal 3-max -> full 3x3 window max
            const float m0 = fmaxf(fmaxf(vml, vm0), vm1);
            const float m1 = fmaxf(fmaxf(vm0, vm1), vm2);
            const float m2 = fmaxf(fmaxf(vm1, vm2), vm3);
            const float m3 = fmaxf(fmaxf(vm2, vm3), vmr);

            acc += loss_elem(p4.x, b4.x, m0);
            acc += loss_elem(p4.y, b4.y, m1);
            acc += loss_elem(p4.z, b4.z, m2);
            acc += loss_elem(p4.w, b4.w, m3);
        }
    }

    // ---- wave32 reduction via V_WMMA_F32_16X16X4_F32 ----
    // A (16x4): lane L<16 holds {A[L][0],A[L][1]}, lane L>=16 holds {A[L-16][2],A[L-16][3]}
    // With a = {acc, 0} and B = ones: D[m][n] = acc_m + acc_{m+16} for all n.
    v2f av; av[0] = acc;  av[1] = 0.0f;
    v2f bv; bv[0] = 1.0f; bv[1] = 1.0f;
    v8f cv = {0.0f, 0.0f, 0.0f, 0.0f, 0.0f, 0.0f, 0.0f, 0.0f};
    cv = __builtin_amdgcn_wmma_f32_16x16x4_f32(false, av, false, bv,
                                               (short)0, cv, false, false);
    // lane n (0-15): C vgpr r = D[r][n]; lane n+16: D[8+r][n]. Column sum + xor-16.
    float s = cv[0] + cv[1] + cv[2] + cv[3] + cv[4] + cv[5] + cv[6] + cv[7];
    s += __shfl_xor(s, 16, 32);

    // ---- block reduction (fixed order -> deterministic) ----
    __shared__ float wsum[TPB / 32];
    const int lane = t & 31;
    const int wave = t >> 5;
    if (lane == 0) wsum[wave] = s;
    __syncthreads();
    if (t == 0) {
        float b = 0.0f;
        for (int w = 0; w < TPB / 32; ++w) b += wsum[w];
        partial[blockIdx.x] = b;
    }
}

__global__ void __launch_bounds__(256)
focal_finish(const float* __restrict__ partial, int n, float* __restrict__ out) {
    __shared__ double sm[256];
    const int t = threadIdx.x;
    double s = 0.0;
    for (int i = t; i < n; i += 256) s += (double)partial[i];
    sm[t] = s;
    __syncthreads();
    for (int off = 128; off > 0; off >>= 1) {
        if (t < off) sm[t] += sm[t + off];
        __syncthreads();
    }
    if (t == 0) out[0] = (float)(sm[0] * (1.0 / ((double)HH * (double)WW)));
}

extern "C" void kernel_launch(void* const* d_in, const int* in_sizes, int n_in,
                              void* d_out, int out_size, void* d_ws, size_t ws_size,
                              hipStream_t stream) {
    const float* pred = (const float*)d_in[0];
    const float* targ = (const float*)d_in[1];
    float* out = (float*)d_out;
    float* partial = (float*)d_ws;   // NBLOCKS floats of scratch

    focal_main<<<dim3(NBLOCKS), dim3(TPB), 0, stream>>>(pred, targ, partial);
    focal_finish<<<dim3(1), dim3(256), 0, stream>>>(partial, NBLOCKS, out);
}